// Layer1Attention_16638703304870
// MI455X (gfx1250) — compile-verified
//
#include <hip/hip_runtime.h>

// Problem constants from the reference.
#define DIGIT_POS 15
#define BATCH 32
#define SEQ 16384
#define DMODEL 64

// Scan decomposition: 1024 tiles of 512, each tile = 2 WMMA chunks of 256.
#define TILE 512
#define TPR (SEQ / TILE)              // 32 tiles per row (== wave width, used in pass 2)
#define CHUNKS_PER_TILE (TILE / 256)  // 2

typedef __attribute__((ext_vector_type(2))) float v2f;
typedef __attribute__((ext_vector_type(8))) float v8f;

__device__ __forceinline__ float lane_shfl(float v, int src) {
  return __shfl(v, src, 32);
}

// D = A(16x4) * B(4x16) + C(16x16), all fp32.  Mirrors the probe-confirmed
// 8-arg pattern of the 16x16x32 f16 builtin.
__device__ __forceinline__ v8f wmma_16x16x4_f32(v2f a, v2f b, v8f c) {
  return __builtin_amdgcn_wmma_f32_16x16x4_f32(
      /*neg_a=*/false, a, /*neg_b=*/false, b,
      /*c_mod=*/(short)0, c, /*reuse_a=*/false, /*reuse_b=*/false);
}

// ---------------- Pass 1: per-tile sums of v = h[...,15]*w + bias ----------
__global__ void pass1_tile_sums(const float* __restrict__ h,
                                const float* __restrict__ vw,
                                const float* __restrict__ vb,
                                float* __restrict__ partial) {
  const int bid  = blockIdx.x;                // 0 .. BATCH*TPR-1
  const int b    = bid / TPR;
  const int tile = bid % TPR;
  const int lane = threadIdx.x & 31;
  const float w = vw[0], bias = vb[0];
  const float* hrow = h + ((size_t)b * SEQ) * DMODEL + DIGIT_POS;
  const int s0 = tile * TILE;

  float sum = 0.f;
  #pragma unroll
  for (int k = 0; k < TILE / 32; ++k) {
    const int s = s0 + lane + 32 * k;
    sum += hrow[(size_t)s * DMODEL] * w + bias;
  }
  #pragma unroll
  for (int d = 16; d; d >>= 1) sum += __shfl_xor(sum, d, 32);
  if (lane == 0) partial[bid] = sum;
}

// ---------------- Pass 2: exclusive scan of the TPR(=32) tile sums per row --
__global__ void pass2_scan_partials(float* __restrict__ partial) {
  const int b    = blockIdx.x;
  const int lane = threadIdx.x & 31;          // lane == tile index (TPR == 32)
  const float v = partial[b * TPR + lane];
  float incl = v;
  #pragma unroll
  for (int d = 1; d < 32; d <<= 1) {
    const int src = (lane >= d) ? (lane - d) : lane;
    const float up = lane_shfl(incl, src);
    if (lane >= d) incl += up;
  }
  partial[b * TPR + lane] = incl - v;         // exclusive prefix
}

// ---------------- Pass 3: WMMA 256-wide scans + carry, divide, store -------
// Chunk layout: X[r][c] = v[s0 + 16*c + r]  (16x16, fp32).
// Y = L*X gives the inclusive scan of each 16-element column; cross-column
// offsets come from a 16-lane shuffle scan of the column totals (row 15).
__global__ void pass3_wmma_scan(const float* __restrict__ h,
                                const float* __restrict__ vw,
                                const float* __restrict__ vb,
                                const float* __restrict__ partial,
                                float* __restrict__ out) {
  const int bid  = blockIdx.x;
  const int b    = bid / TPR;
  const int tile = bid % TPR;
  const int lane = threadIdx.x & 31;
  const int n    = lane & 15;                 // B/C/D column, A row
  const int hi   = lane >> 4;                 // lane half
  const float w = vw[0], bias = vb[0];
  const float* hrow = h + ((size_t)b * SEQ) * DMODEL + DIGIT_POS;
  float* orow = out + (size_t)b * SEQ;
  float carry = partial[bid];

  // Four 16x4 panels of lower-triangular ones L (A operand, f32 layout:
  // lanes 0-15 hold K=0,1 in the two VGPRs, lanes 16-31 hold K=2,3).
  v2f Lp[4];
  #pragma unroll
  for (int p = 0; p < 4; ++p) {
    const int j0 = 4 * p + 2 * hi;            // global column of L
    Lp[p].x = (j0     <= n) ? 1.f : 0.f;
    Lp[p].y = (j0 + 1 <= n) ? 1.f : 0.f;
  }

  int s0 = tile * TILE;
  for (int c = 0; c < CHUNKS_PER_TILE; ++c, s0 += 256) {
    // Prefetch next chunk's (one element per 128B line) while we compute.
    if (c + 1 < CHUNKS_PER_TILE) {
      #pragma unroll
      for (int k = 0; k < 8; ++k)
        __builtin_prefetch(hrow + (size_t)(s0 + 256 + lane + 32 * k) * DMODEL, 0, 3);
    }

    v8f acc = {0.f, 0.f, 0.f, 0.f, 0.f, 0.f, 0.f, 0.f};
    #pragma unroll
    for (int p = 0; p < 4; ++p) {
      const int r0 = 4 * p + 2 * hi;          // rows of X this lane supplies
      v2f bp;
      bp.x = hrow[(size_t)(s0 + 16 * n + r0    ) * DMODEL] * w + bias;
      bp.y = hrow[(size_t)(s0 + 16 * n + r0 + 1) * DMODEL] * w + bias;
      acc = wmma_16x16x4_f32(Lp[p], bp, acc); // Y += L_p * X_p
    }

    // Column totals live at row M=15 -> VGPR 7, lanes 16..31.
    const float t = lane_shfl(acc[7], 16 + n);
    float incl = t;
    #pragma unroll
    for (int d = 1; d < 16; d <<= 1) {
      const int src = (n >= d) ? (lane - d) : lane;
      const float up = lane_shfl(incl, src);
      if (n >= d) incl += up;
    }
    const float exoff = (incl - t) + carry;   // offset for this lane's column
    const float chunk_total = lane_shfl(incl, 31);

    #pragma unroll
    for (int i = 0; i < 8; ++i) {
      const int s = s0 + 16 * n + i + 8 * hi; // C/D layout: M = i + 8*hi, N = n
      orow[s] = (acc[i] + exoff) / (float)(s + 2);
    }
    carry += chunk_total;
  }
}

extern "C" void kernel_launch(void* const* d_in, const int* in_sizes, int n_in,
                              void* d_out, int out_size, void* d_ws, size_t ws_size,
                              hipStream_t stream) {
  (void)in_sizes; (void)n_in; (void)out_size; (void)ws_size;
  const float* h  = (const float*)d_in[0];
  const float* vw = (const float*)d_in[1];
  const float* vb = (const float*)d_in[2];
  float* out      = (float*)d_out;
  float* partial  = (float*)d_ws;             // BATCH*TPR = 1024 floats

  pass1_tile_sums   <<<BATCH * TPR, 32, 0, stream>>>(h, vw, vb, partial);
  pass2_scan_partials<<<BATCH,      32, 0, stream>>>(partial);
  pass3_wmma_scan   <<<BATCH * TPR, 32, 0, stream>>>(h, vw, vb, partial, out);
}